// RNN_38611755991321
// MI455X (gfx1250) — compile-verified
//
#include <hip/hip_runtime.h>

// ---------------------------------------------------------------------------
// Sizes fixed by the reference
// ---------------------------------------------------------------------------
#define B_SZ   256
#define T_SZ   2048
#define F_TR   128
#define F_ST   64
#define H_SZ   3
#define OUT_SZ 32

typedef float v2f __attribute__((ext_vector_type(2)));
typedef float v8f __attribute__((ext_vector_type(8)));

// LDS tile: 16 rows x 128 floats, padded to 132 floats/row so the WMMA
// fragment gather (ds_load_b64, lane L -> row L%16, dword 4*(L%16)+koff)
// hits 64 distinct banks -> conflict-free.
#define ROW_PAD_F   132
#define TILE_F      (16 * ROW_PAD_F)          // floats per wave tile (8448 B)
#define WAVES_PB    4                         // 4 waves/block -> 33792 B LDS

// ---------------------------------------------------------------------------
// Kernel A: xW[t][b][0..2] = tr_input[b][t][:] . W_ih[h][:] + b_ih[h]
// One wave = one 16-row tile (16 consecutive t within one b).
//  1) DMA the 8 KB A-tile global->LDS with 16x global_load_async_to_lds_b128
//     (each instruction = one fully-contiguous 512 B row; ASYNCcnt-tracked,
//     no VGPR round trip).
//  2) s_wait_asynccnt 0 (per-wave: each wave consumes only its own slice).
//  3) 32x { ds_load_b64 A-frag, global_load_b64 W-frag (L0-hot, zero-masked
//     for cols>=3 to keep EXEC all-ones), v_wmma_f32_16x16x4_f32 }.
//  4) Scatter 3 valid output columns to xw[t][b][n] (+bias), transposed
//     [T][B][4] so the scan kernel reads coalesced float4s.
// ---------------------------------------------------------------------------
__global__ void __launch_bounds__(WAVES_PB * 32)
xw_wmma_kernel(const float* __restrict__ tr,
               const float* __restrict__ W_ih,
               const float* __restrict__ b_ih,
               float* __restrict__ xw) {
  __shared__ float smem[WAVES_PB * TILE_F];

  const int lane = threadIdx.x & 31;
  const int wv   = threadIdx.x >> 5;
  const int tile = blockIdx.x * WAVES_PB + wv;   // 32768 tiles total
  const size_t row0 = (size_t)tile * 16;         // flat (b*T + t) of tile row 0
  const int bb = (int)(row0 >> 11);              // T = 2048
  const int t0 = (int)(row0 & (T_SZ - 1));       // multiple of 16

  float* sw = smem + wv * TILE_F;
  // Low 32 bits of the generic (shared-aperture) address == LDS byte offset.
  const unsigned ldsBase = (unsigned)(uintptr_t)sw;

  // ---- 1) async DMA: row i -> LDS row i (padded stride), 512 B / instr ----
  const unsigned gofs0 = (unsigned)(row0 * F_TR * 4) + (unsigned)lane * 16u;
  const unsigned lofs0 = ldsBase + (unsigned)lane * 16u;
#pragma unroll
  for (int i = 0; i < 16; ++i) {
    unsigned l = lofs0 + (unsigned)i * (ROW_PAD_F * 4);
    unsigned g = gofs0 + (unsigned)i * (F_TR * 4);
    asm volatile("global_load_async_to_lds_b128 %0, %1, %2 offset:0"
                 :: "v"(l), "v"(g), "s"(tr)
                 : "memory");
  }
  // ---- 2) wait for our DMA (per-wave producer == consumer) ----
  asm volatile("s_wait_asynccnt 0x0" ::: "memory");

  // ---- 3) WMMA main loop ----
  const int nn = lane & 15;                      // A row-in-tile / B,C column
  const int kh = (lane >> 4) << 1;               // K sub-offset: 0 or 2
  const float* arow = sw + nn * ROW_PAD_F + kh;  // LDS, conflict-free b64 reads

  const int   wrow  = (nn < H_SZ) ? nn : 0;
  const float wmask = (nn < H_SZ) ? 1.0f : 0.0f;
  const float* wr   = W_ih + wrow * F_TR + kh;

  v8f acc = {};
#pragma unroll 8
  for (int k0 = 0; k0 < F_TR; k0 += 4) {
    v2f a = *(const v2f*)(arow + k0);            // ds_load_b64
    v2f w = *(const v2f*)(wr + k0);              // global_load_b64 (L0-hot)
    w.x *= wmask;
    w.y *= wmask;
    acc = __builtin_amdgcn_wmma_f32_16x16x4_f32(
        /*neg_a=*/false, a, /*neg_b=*/false, w,
        /*c_mod=*/(short)0, acc, /*reuse_a=*/false, /*reuse_b=*/false);
  }

  // ---- 4) scatter the 3 valid columns (+bias) ----
  if (nn < H_SZ) {
    const float bias  = b_ih[nn];
    const int   mbase = (lane >> 4) << 3;        // 0 or 8
    float* o = xw + ((size_t)t0 * B_SZ + bb) * 4 + nn;
#pragma unroll
    for (int j = 0; j < 8; ++j) {
      o[(size_t)(mbase + j) * (B_SZ * 4)] = acc[j] + bias;
    }
  }
}

// ---------------------------------------------------------------------------
// Kernel B: per-batch sequential relu-RNN scan + time mean.
// One thread per batch element; each step reads one coalesced float4 from the
// [T][B][4] intermediate (L2-resident). Depth-3 FMA chains per component.
// ---------------------------------------------------------------------------
__global__ void __launch_bounds__(32)
scan_kernel(const float* __restrict__ xw,
            const float* __restrict__ W_hh,
            const float* __restrict__ b_hh,
            float* __restrict__ hmean) {
  const int b = blockIdx.x * blockDim.x + threadIdx.x;
  if (b >= B_SZ) return;

  const float w00 = W_hh[0], w01 = W_hh[1], w02 = W_hh[2];
  const float w10 = W_hh[3], w11 = W_hh[4], w12 = W_hh[5];
  const float w20 = W_hh[6], w21 = W_hh[7], w22 = W_hh[8];
  const float bb0 = b_hh[0], bb1 = b_hh[1], bb2 = b_hh[2];

  float h0 = 0.f, h1 = 0.f, h2 = 0.f;
  float s0 = 0.f, s1 = 0.f, s2 = 0.f;

  const float4* xp = (const float4*)xw;
#pragma unroll 8
  for (int t = 0; t < T_SZ; ++t) {
    // Stream-ahead hint (global_prefetch_b8); speculative, OOB is dropped.
    int tp = t + 64;
    if (tp < T_SZ) __builtin_prefetch(&xp[(size_t)tp * B_SZ + b], 0, 0);

    float4 x = xp[(size_t)t * B_SZ + b];
    float n0 = fmaf(h2, w02, fmaf(h1, w01, fmaf(h0, w00, x.x + bb0)));
    float n1 = fmaf(h2, w12, fmaf(h1, w11, fmaf(h0, w10, x.y + bb1)));
    float n2 = fmaf(h2, w22, fmaf(h1, w21, fmaf(h0, w20, x.z + bb2)));
    h0 = fmaxf(n0, 0.f);
    h1 = fmaxf(n1, 0.f);
    h2 = fmaxf(n2, 0.f);
    s0 += h0; s1 += h1; s2 += h2;
  }

  const float inv = 1.0f / (float)T_SZ;
  hmean[b * 4 + 0] = s0 * inv;
  hmean[b * 4 + 1] = s1 * inv;
  hmean[b * 4 + 2] = s2 * inv;
  hmean[b * 4 + 3] = 0.f;
}

// ---------------------------------------------------------------------------
// Kernel C: out[b][o] = b_fc[o] + st[b][:].W_fc[o][:64] + hmean[b][:].W_fc[o][64:67]
// ---------------------------------------------------------------------------
__global__ void __launch_bounds__(256)
fc_kernel(const float* __restrict__ st,
          const float* __restrict__ hmean,
          const float* __restrict__ W_fc,
          const float* __restrict__ b_fc,
          float* __restrict__ out) {
  const int idx = blockIdx.x * blockDim.x + threadIdx.x;   // 0..8191
  const int b = idx >> 5;
  const int o = idx & 31;

  const float* w = W_fc + o * (F_ST + H_SZ);
  const float* s = st + b * F_ST;
  float acc = b_fc[o];
#pragma unroll 8
  for (int f = 0; f < F_ST; ++f) acc = fmaf(s[f], w[f], acc);

  const float* hm = hmean + b * 4;
  acc = fmaf(hm[0], w[F_ST + 0], acc);
  acc = fmaf(hm[1], w[F_ST + 1], acc);
  acc = fmaf(hm[2], w[F_ST + 2], acc);

  out[idx] = acc;   // idx == b*32 + o
}

// ---------------------------------------------------------------------------
// Launch
// ---------------------------------------------------------------------------
extern "C" void kernel_launch(void* const* d_in, const int* in_sizes, int n_in,
                              void* d_out, int out_size, void* d_ws, size_t ws_size,
                              hipStream_t stream) {
  (void)in_sizes; (void)n_in; (void)out_size; (void)ws_size;

  const float* tr   = (const float*)d_in[0];  // [B,T,F_TR]
  const float* st   = (const float*)d_in[1];  // [B,F_ST]
  const float* W_ih = (const float*)d_in[2];  // [H,F_TR]
  const float* W_hh = (const float*)d_in[3];  // [H,H]
  const float* b_ih = (const float*)d_in[4];  // [H]
  const float* b_hh = (const float*)d_in[5];  // [H]
  const float* W_fc = (const float*)d_in[6];  // [OUT,H+F_ST]
  const float* b_fc = (const float*)d_in[7];  // [OUT]
  float* out = (float*)d_out;                 // [B,OUT]

  // Workspace: xw [T][B][4] = 8 MB, then hmean [B][4].
  float* xw    = (float*)d_ws;
  float* hmean = (float*)((char*)d_ws + (size_t)T_SZ * B_SZ * 4 * sizeof(float));

  // Phase 1: 32768 tiles, 4 waves (tiles) per 128-thread block.
  xw_wmma_kernel<<<(B_SZ * T_SZ / 16) / WAVES_PB, WAVES_PB * 32, 0, stream>>>(
      tr, W_ih, b_ih, xw);

  // Phase 2: 256 sequential scans, spread over 8 WGPs.
  scan_kernel<<<8, 32, 0, stream>>>(xw, W_hh, b_hh, hmean);

  // Phase 3: tiny FC, 8192 threads.
  fc_kernel<<<32, 256, 0, stream>>>(st, hmean, W_fc, b_fc, out);
}